// MPC_56856777064682
// MI455X (gfx1250) — compile-verified
//
#include <hip/hip_runtime.h>

#define T_  100
#define B_  4096
#define NS_ 12
#define NC_ 4
#define N_  16

typedef float v2f __attribute__((ext_vector_type(2)));
typedef float v8f __attribute__((ext_vector_type(8)));

// One 16x16x4 fp32 WMMA step: D = A(16x4) * B(4x16) + C
static __device__ __forceinline__ v8f wmma4(v2f a, v2f b, v8f c) {
  return __builtin_amdgcn_wmma_f32_16x16x4_f32(
      /*neg_a=*/false, a, /*neg_b=*/false, b,
      /*c_mod=*/(short)0, c, /*reuse_a=*/false, /*reuse_b=*/false);
}

// Layouts (wave32, CDNA5 ISA 7.12.2), h = lane>>4, col = lane&15:
//   C/D (8 vgprs):   Cc[r]   = M[r + 8*h][col]
//   A   (2 vgprs/k): A[k][p] = M[lane&15][4k + 2h + p]   (M x K)
//   B   (2 vgprs/k): B[k][p] = M[4k + 2h + p][col]       (K x N)
// C->B conversion: with Gx[r] = shfl_xor(Gc[r],16) (partner half's rows),
// all selections use compile-time register indices:
//   chunk0 = h ? {Gx2,Gx3} : {Gc0,Gc1}
//   chunk1 = h ? {Gx6,Gx7} : {Gc4,Gc5}
//   chunk2 = h ? {Gc2,Gc3} : {Gx0,Gx1}
// (chunk3 = rows 12..15 is never needed: F/V are zero there.)

__global__ __launch_bounds__(128) void lqr_kernel(
    const float* __restrict__ xinit,  // (B, NS)
    const float* __restrict__ Cm,     // (T, B, N, N)
    const float* __restrict__ cvec,   // (T, B, N)
    const float* __restrict__ Fm,     // (T-1, B, NS, N)
    const float* __restrict__ fvec,   // (T-1, B, NS)
    float* __restrict__ out,          // xs | us | costs
    float* __restrict__ Kbuf,         // (T, B, NC, NS) scratch
    float* __restrict__ kbuf)         // (T, B, NC)     scratch
{
  const int lane = threadIdx.x & 31;
  const int w    = threadIdx.x >> 5;
  const int b    = blockIdx.x * 4 + w;
  const int h    = lane >> 4;
  const int col  = lane & 15;
  const bool hb  = (h != 0);

  __shared__ float sQ [4][N_ * N_];   // Q_t (full 16x16)
  __shared__ float sVn[4][N_ * N_];   // unsymmetrized V'
  __shared__ float sVs[4][N_ * N_];   // symmetrized V (row-major, 12x12 live)
  __shared__ float sVf[4][N_];        // V f_t + v
  __shared__ float sv [4][N_];        // v
  __shared__ float sq [4][N_];        // q
  __shared__ float sK [4][NC_ * NS_];
  __shared__ float sk [4][NC_];

  // init: V = 0, v = 0
  #pragma unroll
  for (int r = 0; r < 8; ++r) sVs[w][(r + 8 * h) * N_ + col] = 0.f;
  if (lane < N_) sv[w][lane] = 0.f;

  // ---------------- backward Riccati pass ----------------
  for (int t = T_ - 1; t >= 0; --t) {
    const bool   hasF = (t < T_ - 1);
    const size_t tb   = (size_t)t * B_ + b;
    const float* Ft   = Fm   + tb * (NS_ * N_);
    const float* Ct   = Cm   + tb * (N_ * N_);
    const float* ct   = cvec + tb * N_;

    // F_t in B-operand layout, chunks 0..2 (rows 0..11; chunk3 rows are 0).
    // A-layout of F^T is elementwise identical to this. F is read ONCE here;
    // the q-step below reuses these registers.
    float Fb[3][2];
    #pragma unroll
    for (int k = 0; k < 3; ++k)
      #pragma unroll
      for (int p = 0; p < 2; ++p) {
        int K = 4 * k + 2 * h + p;
        Fb[k][p] = hasF ? Ft[K * N_ + col] : 0.f;
      }

    // wave-uniform f_t (scalar loads)
    float fts[NS_];
    #pragma unroll
    for (int s = 0; s < NS_; ++s) fts[s] = hasF ? fvec[tb * NS_ + s] : 0.f;

    __syncthreads();  // B0: prev iteration done with LDS; sVs/sv ready

    // A-operand of V straight from LDS (symmetry: V[col][K] == V[K][col];
    // reading row-K form is LDS-bank-conflict-free)
    float Va[3][2];
    #pragma unroll
    for (int k = 0; k < 3; ++k)
      #pragma unroll
      for (int p = 0; p < 2; ++p)
        Va[k][p] = sVs[w][(4 * k + 2 * h + p) * N_ + col];

    // Vf = V f_t + v  (half-0 lanes; symmetric read keeps banks clean)
    if (h == 0) {
      float acc = sv[w][col];
      if (hasF) {
        #pragma unroll
        for (int s = 0; s < NS_; ++s) acc += sVs[w][s * N_ + col] * fts[s];
      }
      sVf[w][col] = acc;
    }

    // G = V * F   (3 chained f32 WMMAs; K-chunk 3 is all zero)
    v8f G;
    #pragma unroll
    for (int r = 0; r < 8; ++r) G[r] = 0.f;
    #pragma unroll
    for (int k = 0; k < 3; ++k) {
      v2f a = {Va[k][0], Va[k][1]};
      v2f bm = {Fb[k][0], Fb[k][1]};
      G = wmma4(a, bm, G);
    }

    // G (C-layout) -> B-operand layout: 6 shuffles + 6 constant-index selects
    float gx0 = __shfl_xor((float)G[0], 16, 32);
    float gx1 = __shfl_xor((float)G[1], 16, 32);
    float gx2 = __shfl_xor((float)G[2], 16, 32);
    float gx3 = __shfl_xor((float)G[3], 16, 32);
    float gx6 = __shfl_xor((float)G[6], 16, 32);
    float gx7 = __shfl_xor((float)G[7], 16, 32);
    v2f Gb0 = {hb ? gx2 : (float)G[0], hb ? gx3 : (float)G[1]};
    v2f Gb1 = {hb ? gx6 : (float)G[4], hb ? gx7 : (float)G[5]};
    v2f Gb2 = {hb ? (float)G[2] : gx0, hb ? (float)G[3] : gx1};

    // Q = C_t + F^T * G   (accumulate onto loaded C_t; chunk 3 zero)
    v8f Q;
    #pragma unroll
    for (int r = 0; r < 8; ++r) Q[r] = Ct[(r + 8 * h) * N_ + col];
    {
      v2f a0 = {Fb[0][0], Fb[0][1]};
      v2f a1 = {Fb[1][0], Fb[1][1]};
      v2f a2 = {Fb[2][0], Fb[2][1]};
      Q = wmma4(a0, Gb0, Q);
      Q = wmma4(a1, Gb1, Q);
      Q = wmma4(a2, Gb2, Q);
    }

    #pragma unroll
    for (int r = 0; r < 8; ++r) sQ[w][(r + 8 * h) * N_ + col] = Q[r];

    __syncthreads();  // B1: sQ, sVf visible

    // q = c_t + F^T * Vf, reusing Fb registers (each lane-half holds a
    // complementary set of 6 rows of F; one xor-16 shuffle completes the sum)
    {
      float part = 0.f;
      #pragma unroll
      for (int k = 0; k < 3; ++k)
        #pragma unroll
        for (int p = 0; p < 2; ++p)
          part += Fb[k][p] * sVf[w][4 * k + 2 * h + p];
      part += __shfl_xor(part, 16, 32);
      if (h == 0) sq[w][col] = ct[col] + part;
    }
    __syncthreads();  // B2: sq visible

    // Kk = -Quu^{-1} [Qux | qu] : 13 lanes, one RHS column each.
    // Quu is SPD -> Gaussian elimination without pivoting.
    if (lane < 13) {
      float A[4][4], xr[4];
      #pragma unroll
      for (int r = 0; r < 4; ++r) {
        #pragma unroll
        for (int c2 = 0; c2 < 4; ++c2)
          A[r][c2] = sQ[w][(NS_ + r) * N_ + NS_ + c2];
        xr[r] = (lane < NS_) ? sQ[w][(NS_ + r) * N_ + lane] : sq[w][NS_ + r];
      }
      #pragma unroll
      for (int kk = 0; kk < 4; ++kk) {
        float inv = 1.f / A[kk][kk];
        #pragma unroll
        for (int rr = kk + 1; rr < 4; ++rr) {
          float m = A[rr][kk] * inv;
          #pragma unroll
          for (int c2 = kk + 1; c2 < 4; ++c2) A[rr][c2] -= m * A[kk][c2];
          xr[rr] -= m * xr[kk];
        }
      }
      #pragma unroll
      for (int kk = 3; kk >= 0; --kk) {
        float s2 = xr[kk];
        #pragma unroll
        for (int c2 = kk + 1; c2 < 4; ++c2) s2 -= A[kk][c2] * xr[c2];
        xr[kk] = s2 / A[kk][kk];
      }
      if (lane < NS_) {
        #pragma unroll
        for (int r = 0; r < 4; ++r) {
          float kv = -xr[r];
          sK[w][r * NS_ + lane] = kv;
          Kbuf[(tb * NC_ + r) * NS_ + lane] = kv;
        }
      } else {
        #pragma unroll
        for (int r = 0; r < 4; ++r) {
          float kv = -xr[r];
          sk[w][r] = kv;
          kbuf[tb * NC_ + r] = kv;
        }
      }
    }
    __syncthreads();  // B3: sK, sk visible

    // Vn = Qxx + Qxu*K (zero-padded), vn = qx + Qxu*k
    #pragma unroll
    for (int r = 0; r < 8; ++r) {
      int i = r + 8 * h;
      float val = 0.f;
      if (i < NS_ && col < NS_) {
        val = sQ[w][i * N_ + col];
        #pragma unroll
        for (int u = 0; u < NC_; ++u)
          val += sQ[w][i * N_ + NS_ + u] * sK[w][u * NS_ + col];
      }
      sVn[w][i * N_ + col] = val;
    }
    if (h == 0) {
      float vnv = 0.f;
      if (col < NS_) {
        vnv = sq[w][col];
        #pragma unroll
        for (int u = 0; u < NC_; ++u)
          vnv += sQ[w][col * N_ + NS_ + u] * sk[w][u];
      }
      sv[w][col] = vnv;
    }
    __syncthreads();  // B4: sVn visible

    // V = 0.5*(Vn + Vn^T) -> back into sVs for next iteration
    #pragma unroll
    for (int r = 0; r < 8; ++r) {
      int i = r + 8 * h;
      sVs[w][i * N_ + col] =
          0.5f * (sVn[w][i * N_ + col] + sVn[w][col * N_ + i]);
    }
  }

  // ---------------- forward rollout ----------------
  float* xs_out  = out;
  float* us_out  = out + (size_t)T_ * B_ * NS_;
  float* obj_out = out + (size_t)T_ * B_ * NS_ + (size_t)T_ * B_ * NC_;

  float costAcc = 0.f;
  __syncthreads();
  if (lane < N_)
    sVf[w][lane] = (lane < NS_) ? xinit[(size_t)b * NS_ + lane] : 0.f;
  __syncthreads();

  for (int t = 0; t < T_; ++t) {
    const bool   hasF = (t < T_ - 1);
    const size_t tb   = (size_t)t * B_ + b;
    const float* Ct   = Cm   + tb * (N_ * N_);
    const float* ct   = cvec + tb * N_;

    // tau = [x; u], u = K x + k ; emit xs, us
    if (lane < NS_) {
      float xv = sVf[w][lane];
      sVn[w][lane] = xv;
      xs_out[tb * NS_ + lane] = xv;
    } else if (lane < N_) {
      int r = lane - NS_;
      float uv = kbuf[tb * NC_ + r];
      #pragma unroll
      for (int j = 0; j < NS_; ++j)
        uv += Kbuf[(tb * NC_ + r) * NS_ + j] * sVf[w][j];
      sVn[w][lane] = uv;
      us_out[tb * NC_ + r] = uv;
    }
    __syncthreads();

    // cost = 0.5*tau^T C tau + c.tau  (half-0 lanes, full-wave reduce)
    float partc = 0.f;
    if (h == 0) {
      float csum = 0.f;
      #pragma unroll
      for (int i = 0; i < N_; ++i) csum += Ct[i * N_ + col] * sVn[w][i];
      partc = sVn[w][col] * (0.5f * csum + ct[col]);
    }
    #pragma unroll
    for (int s = 1; s < 32; s <<= 1) partc += __shfl_xor(partc, s, 32);
    costAcc += partc;

    // x_{t+1} = F tau + f
    float xn = 0.f;
    if (hasF && lane < NS_) {
      const float* Ft = Fm + tb * (NS_ * N_);
      xn = fvec[tb * NS_ + lane];
      #pragma unroll
      for (int j = 0; j < N_; ++j) xn += Ft[lane * N_ + j] * sVn[w][j];
    }
    __syncthreads();               // sVn reads done before sVf rewrite
    if (lane < N_) sVf[w][lane] = (lane < NS_) ? xn : 0.f;
    __syncthreads();
  }

  if (lane == 0) obj_out[b] = costAcc;
}

extern "C" void kernel_launch(void* const* d_in, const int* in_sizes, int n_in,
                              void* d_out, int out_size, void* d_ws, size_t ws_size,
                              hipStream_t stream) {
  (void)in_sizes; (void)n_in; (void)out_size; (void)ws_size;
  const float* xinit = (const float*)d_in[0];
  const float* Cm    = (const float*)d_in[1];
  const float* cvec  = (const float*)d_in[2];
  const float* Fm    = (const float*)d_in[3];
  const float* fvec  = (const float*)d_in[4];
  float* out  = (float*)d_out;
  float* Kbuf = (float*)d_ws;                                  // T*B*NC*NS
  float* kbuf = Kbuf + (size_t)T_ * B_ * NC_ * NS_;            // T*B*NC

  dim3 grid(B_ / 4), block(128);
  lqr_kernel<<<grid, block, 0, stream>>>(xinit, Cm, cvec, Fm, fvec,
                                         out, Kbuf, kbuf);
}